// GraphCovLayer_11519102287947
// MI455X (gfx1250) — compile-verified
//
#include <hip/hip_runtime.h>

#define N_USERS 50000
#define N_ITEMS 20000
#define N_EDGES 1000000
#define N_RAT   5
#define FEAT    64

#define U_TILES (N_USERS / 16)   // 3125
#define V_TILES (N_ITEMS / 16)   // 1250
#define WPB     4                // waves per block in gemm kernel

typedef __attribute__((ext_vector_type(2))) float v2f;
typedef __attribute__((ext_vector_type(8))) float v8f;

// ---------------- Phase 0: zero the segment-sum / count scratch ----------------
__global__ void zero_ws(float* __restrict__ p, long n) {
    long i      = (long)blockIdx.x * blockDim.x + threadIdx.x;
    long stride = (long)gridDim.x * blockDim.x;
    for (; i < n; i += stride) p[i] = 0.0f;
}

// ---------------- Phase 1: edge scatter (segment sums + counts) ----------------
// One wave32 per edge; lane i handles features {2i, 2i+1} (float2, coalesced).
__global__ __launch_bounds__(256) void scatter_edges(
    const int*   __restrict__ u_s, const int* __restrict__ v_s,
    const int*   __restrict__ rate,
    const float* __restrict__ x_user, const float* __restrict__ x_item,
    float* __restrict__ sum_u, float* __restrict__ cnt_u,
    float* __restrict__ sum_v, float* __restrict__ cnt_v)
{
    int tid  = blockIdx.x * blockDim.x + threadIdx.x;
    int e    = tid >> 5;
    int lane = tid & 31;
    if (e >= N_EDGES) return;

    int u = u_s[e];
    int v = v_s[e];
    int r = rate[e];

    float2 xi = ((const float2*)(x_item + (size_t)v * FEAT))[lane];
    float2 xu = ((const float2*)(x_user + (size_t)u * FEAT))[lane];

    float* du = sum_u + ((size_t)u * N_RAT + r) * FEAT + 2 * lane;
    float* dv = sum_v + ((size_t)v * N_RAT + r) * FEAT + 2 * lane;
    atomicAdd(du,     xi.x);
    atomicAdd(du + 1, xi.y);
    atomicAdd(dv,     xu.x);
    atomicAdd(dv + 1, xu.y);

    if (lane == 0) {
        atomicAdd(cnt_u + (size_t)u * N_RAT + r, 1.0f);
        atomicAdd(cnt_v + (size_t)v * N_RAT + r, 1.0f);
    }
}

// ---------------- Phase 2: mean + rating-specific GEMM via f32 WMMA ----------------
// One wave per 16-node tile. D(16x64) = sum_r mean_r(16x64) x W[r](64x64),
// built from V_WMMA_F32_16X16X4_F32 (M=16,N=16,K=4), 4 N-tiles, K stepped by 4.
__global__ __launch_bounds__(WPB * 32) void gemm_out(
    const float* __restrict__ sum_u, const float* __restrict__ cnt_u,
    const float* __restrict__ sum_v, const float* __restrict__ cnt_v,
    const float* __restrict__ weight, float* __restrict__ out)
{
    __shared__ float sW[N_RAT * FEAT * FEAT];   // 80 KB of the 320 KB WGP LDS
    for (int i = threadIdx.x; i < N_RAT * FEAT * FEAT; i += blockDim.x)
        sW[i] = weight[i];
    __syncthreads();

    int wave = threadIdx.x >> 5;
    int lane = threadIdx.x & 31;
    int tile = blockIdx.x * WPB + wave;
    if (tile >= U_TILES + V_TILES) return;      // wave-uniform exit

    const float* sum; const float* cnt; float* op; int node_base;
    if (tile < U_TILES) {                       // wave-uniform select
        sum = sum_u; cnt = cnt_u; op = out;
        node_base = tile * 16;
    } else {
        sum = sum_v; cnt = cnt_v; op = out + (size_t)N_USERS * FEAT;
        node_base = (tile - U_TILES) * 16;
    }

    int mrow = lane & 15;     // A: row M for this lane; B/C/D: column N
    int half = lane >> 4;     // 0: K={k0,k0+1}; 1: K={k0+2,k0+3}
    int node = node_base + mrow;

    v8f acc0 = {}, acc1 = {}, acc2 = {}, acc3 = {};

    for (int r = 0; r < N_RAT; ++r) {
        float c  = cnt[(size_t)node * N_RAT + r];
        float rc = 1.0f / fmaxf(c, 1.0f);       // mean = sum / max(cnt,1)
        const float* srow = sum + ((size_t)node * N_RAT + r) * FEAT;
        const float* wr   = sW + r * FEAT * FEAT;

        #pragma unroll
        for (int k0 = 0; k0 < FEAT; k0 += 4) {
            int k = k0 + 2 * half;
            // A fragment (16x4 f32): lane holds mean[node][k], mean[node][k+1]
            float2 av = *(const float2*)(srow + k);
            v2f a; a.x = av.x * rc; a.y = av.y * rc;
            // B fragments (4x16 f32) from LDS: lane holds W[k][n], W[k+1][n]
            const float* wk = wr + k * FEAT + mrow;
            v2f b0, b1, b2, b3;
            b0.x = wk[0];  b0.y = wk[FEAT];
            b1.x = wk[16]; b1.y = wk[FEAT + 16];
            b2.x = wk[32]; b2.y = wk[FEAT + 32];
            b3.x = wk[48]; b3.y = wk[FEAT + 48];
            acc0 = __builtin_amdgcn_wmma_f32_16x16x4_f32(false, a, false, b0, (short)0, acc0, false, false);
            acc1 = __builtin_amdgcn_wmma_f32_16x16x4_f32(false, a, false, b1, (short)0, acc1, false, false);
            acc2 = __builtin_amdgcn_wmma_f32_16x16x4_f32(false, a, false, b2, (short)0, acc2, false, false);
            acc3 = __builtin_amdgcn_wmma_f32_16x16x4_f32(false, a, false, b3, (short)0, acc3, false, false);
        }
    }

    // C/D layout: lane = N (mod 16), VGPR j = row M=j (lanes 0-15) / M=8+j (16-31)
    #pragma unroll
    for (int j = 0; j < 8; ++j) {
        int m = j + 8 * half;
        float* orow = op + (size_t)(node_base + m) * FEAT + mrow;
        orow[0]  = acc0[j];
        orow[16] = acc1[j];
        orow[32] = acc2[j];
        orow[48] = acc3[j];
    }
}

// ---------------- launcher ----------------
extern "C" void kernel_launch(void* const* d_in, const int* in_sizes, int n_in,
                              void* d_out, int out_size, void* d_ws, size_t ws_size,
                              hipStream_t stream) {
    const int*   u_s    = (const int*)d_in[0];
    const int*   v_s    = (const int*)d_in[1];
    const int*   rate   = (const int*)d_in[2];
    const float* x_user = (const float*)d_in[3];
    const float* x_item = (const float*)d_in[4];
    const float* weight = (const float*)d_in[5];
    float*       out    = (float*)d_out;

    // scratch layout (floats): sum_u | cnt_u | sum_v | cnt_v  (~91 MB total)
    float* sum_u = (float*)d_ws;
    float* cnt_u = sum_u + (size_t)N_USERS * N_RAT * FEAT;
    float* sum_v = cnt_u + (size_t)N_USERS * N_RAT;
    float* cnt_v = sum_v + (size_t)N_ITEMS * N_RAT * FEAT;
    long ws_floats = (long)N_USERS * N_RAT * FEAT + (long)N_USERS * N_RAT
                   + (long)N_ITEMS * N_RAT * FEAT + (long)N_ITEMS * N_RAT;

    zero_ws<<<2048, 256, 0, stream>>>((float*)d_ws, ws_floats);

    int scatter_blocks = (int)(((long)N_EDGES * 32 + 255) / 256);   // 125000
    scatter_edges<<<scatter_blocks, 256, 0, stream>>>(
        u_s, v_s, rate, x_user, x_item, sum_u, cnt_u, sum_v, cnt_v);

    int tiles = U_TILES + V_TILES;                                  // 4375
    gemm_out<<<(tiles + WPB - 1) / WPB, WPB * 32, 0, stream>>>(
        sum_u, cnt_u, sum_v, cnt_v, weight, out);
}